// OctreeMaxUnpool_51677046505683
// MI455X (gfx1250) — compile-verified
//
#include <hip/hip_runtime.h>

// OctreeMaxUnpool for MI455X (gfx1250).
// Pure scatter/gather, 0 FLOPs, ~690 MB of traffic -> HBM-bound (~30 us @ 23.3 TB/s).
// Strategy: 1 thread per (channel, node) group; build the one-hot 8-float group in
// registers and emit two 16-byte non-temporal global stores (write-once 537 MB stream),
// non-temporal loads for the 67 MB mask stream, regular (cached) loads for the sorted
// octree gather (L2-resident, heavy reuse across 128 channels).

#define H_IN   160000
#define H_NE   131072      // 2^17
#define H_NE_LOG2 17
#define C_CH   128
#define H_OUT  (8 * H_NE)  // 1048576

typedef float v4f __attribute__((ext_vector_type(4)));

__global__ __launch_bounds__(256) void
OctreeMaxUnpool_kernel(const float* __restrict__ data,
                       const int*   __restrict__ mask,
                       const int*   __restrict__ octree,
                       float*       __restrict__ out)
{
    const int tid = blockIdx.x * blockDim.x + threadIdx.x;   // 0 .. C*H_NE-1
    const int i   = tid & (H_NE - 1);                        // node index (fast dim -> coalesced)
    const int c   = tid >> H_NE_LOG2;                        // channel

    // octree: 512 KB total, read by all 128 channels -> keep temporal (L2 hit).
    const int oc = octree[i];

    // data gather: octree is sorted, so addresses are near-sequential per channel;
    // lines are reused ~26x within a wavefront -> temporal load.
    const float val = data[c * H_IN + oc];

    // mask: streamed exactly once -> non-temporal.
    const int m = __builtin_nontemporal_load(mask + tid);    // mask flat index == tid

    // Build the one-hot group of 8 in registers (lowers to v_cndmask_b32 x8).
    v4f lo, hi;
    lo.x = (m == 0) ? val : 0.0f;
    lo.y = (m == 1) ? val : 0.0f;
    lo.z = (m == 2) ? val : 0.0f;
    lo.w = (m == 3) ? val : 0.0f;
    hi.x = (m == 4) ? val : 0.0f;
    hi.y = (m == 5) ? val : 0.0f;
    hi.z = (m == 6) ? val : 0.0f;
    hi.w = (m == 7) ? val : 0.0f;

    // 32B-aligned, write-once, 537 MB stream (> 192 MB L2) -> non-temporal b128 stores.
    v4f* dst = (v4f*)(out + (size_t)c * H_OUT + 8u * (unsigned)i);
    __builtin_nontemporal_store(lo, dst);
    __builtin_nontemporal_store(hi, dst + 1);
}

extern "C" void kernel_launch(void* const* d_in, const int* in_sizes, int n_in,
                              void* d_out, int out_size, void* d_ws, size_t ws_size,
                              hipStream_t stream)
{
    const float* data   = (const float*)d_in[0];   // (1, C, H, 1) fp32
    const int*   mask   = (const int*)  d_in[1];   // (C, H_NE) int32 in [0,8)
    const int*   octree = (const int*)  d_in[2];   // (H_NE,) int32, sorted
    float*       out    = (float*)      d_out;     // (1, C, 8*H_NE, 1) fp32

    const int total  = C_CH * H_NE;                // 16,777,216 threads
    const int block  = 256;                        // 8 wave32 per workgroup
    const int grid   = total / block;              // 65,536 blocks

    OctreeMaxUnpool_kernel<<<grid, block, 0, stream>>>(data, mask, octree, out);
}